// PrototypeBank_85134841741643
// MI455X (gfx1250) — compile-verified
//
#include <hip/hip_runtime.h>
#include <math.h>

#define N_CLASSES 1000
#define EMBED_DIM 512
#define BATCH     16384
#define NT        63            // ceil(1000/16)
#define LDS_STRIDE 520          // 512 + 8 floats pad -> avoids 16-way LDS bank conflicts

typedef __attribute__((ext_vector_type(2))) float v2f;
typedef __attribute__((ext_vector_type(8))) float v8f;

// ---------- row-wise L2 normalize; optional transposed output ----------
__global__ void __launch_bounds__(256)
rownorm_kernel(const float* __restrict__ in, float* __restrict__ out,
               int transpose, int out_ld) {
  __shared__ float red[256];
  const int r = blockIdx.x;
  const int t = threadIdx.x;
  const float* row = in + (size_t)r * EMBED_DIM;
  float x0 = row[t];
  float x1 = row[t + 256];
  red[t] = x0 * x0 + x1 * x1;
  __syncthreads();
  for (int s = 128; s > 0; s >>= 1) {
    if (t < s) red[t] += red[t + s];
    __syncthreads();
  }
  float d = fmaxf(sqrtf(red[0]), 1e-12f);
  float y0 = x0 / d;
  float y1 = x1 / d;
  if (transpose) {
    out[(size_t)t * out_ld + r]         = y0;
    out[(size_t)(t + 256) * out_ld + r] = y1;
  } else {
    out[(size_t)r * EMBED_DIM + t]       = y0;
    out[(size_t)r * EMBED_DIM + t + 256] = y1;
  }
}

// ---------- fp32 WMMA GEMM: logits = featsN @ protosN^T / 0.1 ----------
// featsN: [BATCH,512] row-major; protosT: [512,1000] (protosN transposed)
__global__ void __launch_bounds__(256)
gemm_wmma_kernel(const float* __restrict__ featsN,
                 const float* __restrict__ protosT,
                 float* __restrict__ logits) {
  __shared__ float aT[16 * LDS_STRIDE];
  const int tid = threadIdx.x;
  const int rowbase = blockIdx.x * 16;

  // Stage the 16x512 A tile into LDS (float4 granularity, padded rows)
  for (int v = tid; v < 16 * 128; v += 256) {
    const int rr = v >> 7;           // / 128
    const int cc = (v & 127) << 2;   // * 4
    const float4 val = *(const float4*)(featsN + (size_t)(rowbase + rr) * EMBED_DIM + cc);
    float* dst = &aT[rr * LDS_STRIDE + cc];
    dst[0] = val.x; dst[1] = val.y; dst[2] = val.z; dst[3] = val.w;
  }
  __syncthreads();

  const int wave = tid >> 5;
  const int lane = tid & 31;
  const int m  = lane & 15;    // A row / B col within tile
  const int kh = lane >> 4;    // which K-half pair this lane owns
  const float* aRow = &aT[m * LDS_STRIDE + 2 * kh];

  for (int jt = wave; jt < NT; jt += 8) {
    const int col = jt * 16 + m;
    v8f acc = {0.f, 0.f, 0.f, 0.f, 0.f, 0.f, 0.f, 0.f};

    if (jt < NT - 1) {
      // -------- full tile: no bounds masking in the hot loop --------
      const float* bPtr = protosT + (size_t)(2 * kh) * N_CLASSES + col;
#pragma unroll 8
      for (int k0 = 0; k0 < EMBED_DIM; k0 += 4) {
        v2f a;
        a.x = aRow[k0];
        a.y = aRow[k0 + 1];
        v2f b;
        b.x = bPtr[(size_t)k0 * N_CLASSES];
        b.y = bPtr[(size_t)(k0 + 1) * N_CLASSES];
        acc = __builtin_amdgcn_wmma_f32_16x16x4_f32(
            false, a, false, b, (short)0, acc, false, false);
      }
#pragma unroll
      for (int r2 = 0; r2 < 8; ++r2) {
        const int row = rowbase + r2 + 8 * kh;   // D layout: VGPR r -> M = r + 8*kh
        logits[(size_t)row * N_CLASSES + col] = acc[r2] * 10.0f;
      }
    } else {
      // -------- tail tile (cols 992..1007): mask lanes >= N_CLASSES --------
      const bool cok = (col < N_CLASSES);
      const int colc = cok ? col : (N_CLASSES - 1);
      const float* bPtr = protosT + (size_t)(2 * kh) * N_CLASSES + colc;
#pragma unroll 4
      for (int k0 = 0; k0 < EMBED_DIM; k0 += 4) {
        v2f a;
        a.x = aRow[k0];
        a.y = aRow[k0 + 1];
        float bx = bPtr[(size_t)k0 * N_CLASSES];
        float by = bPtr[(size_t)(k0 + 1) * N_CLASSES];
        v2f b;
        b.x = cok ? bx : 0.0f;
        b.y = cok ? by : 0.0f;
        acc = __builtin_amdgcn_wmma_f32_16x16x4_f32(
            false, a, false, b, (short)0, acc, false, false);
      }
      if (cok) {
#pragma unroll
        for (int r2 = 0; r2 < 8; ++r2) {
          const int row = rowbase + r2 + 8 * kh;
          logits[(size_t)row * N_CLASSES + col] = acc[r2] * 10.0f;
        }
      }
    }
  }
}

// ---------- per-row NLL from logits ----------
__global__ void __launch_bounds__(256)
rowloss_kernel(const float* __restrict__ logits, const int* __restrict__ labels,
               float* __restrict__ rowloss) {
  __shared__ float red[256];
  const int r = blockIdx.x;
  const int t = threadIdx.x;
  const float* row = logits + (size_t)r * N_CLASSES;

  float mx = -INFINITY;
  for (int c = t; c < N_CLASSES; c += 256) mx = fmaxf(mx, row[c]);
  red[t] = mx;
  __syncthreads();
  for (int s = 128; s > 0; s >>= 1) {
    if (t < s) red[t] = fmaxf(red[t], red[t + s]);
    __syncthreads();
  }
  mx = red[0];
  __syncthreads();

  float se = 0.0f;
  for (int c = t; c < N_CLASSES; c += 256) se += expf(row[c] - mx);
  red[t] = se;
  __syncthreads();
  for (int s = 128; s > 0; s >>= 1) {
    if (t < s) red[t] += red[t + s];
    __syncthreads();
  }
  if (t == 0) {
    const int lbl = labels[r];
    rowloss[r] = logf(red[0]) + mx - row[lbl];   // = -logp[label]
  }
}

// ---------- final loss mean (deterministic tree reduce) ----------
__global__ void __launch_bounds__(256)
lossreduce_kernel(const float* __restrict__ rowloss, float* __restrict__ out0) {
  __shared__ float red[256];
  const int t = threadIdx.x;
  float s = 0.0f;
  for (int i = t; i < BATCH; i += 256) s += rowloss[i];
  red[t] = s;
  __syncthreads();
  for (int k = 128; k > 0; k >>= 1) {
    if (t < k) red[t] += red[t + k];
    __syncthreads();
  }
  if (t == 0) out0[0] = red[0] / (float)BATCH;
}

// ---------- per-class deterministic segment mean + L2 norm + EMA ----------
__global__ void __launch_bounds__(256)
classupdate_kernel(const float* __restrict__ featsN, const int* __restrict__ labels,
                   const float* __restrict__ prototypes, const float* __restrict__ class_counts,
                   float* __restrict__ outProto, float* __restrict__ outCounts) {
  __shared__ int   lbl[256];
  __shared__ float red[256];
  const int c = blockIdx.x;
  const int t = threadIdx.x;

  float a0 = 0.0f, a1 = 0.0f;
  int cnt = 0;
  for (int base = 0; base < BATCH; base += 256) {
    __syncthreads();
    lbl[t] = labels[base + t];
    __syncthreads();
    for (int j = 0; j < 256; ++j) {
      if (lbl[j] == c) {              // uniform branch within block
        const float* fr = featsN + (size_t)(base + j) * EMBED_DIM;
        a0 += fr[t];
        a1 += fr[t + 256];
        ++cnt;
      }
    }
  }

  const float denom = fmaxf((float)cnt, 1.0f);
  const float m0 = a0 / denom;
  const float m1 = a1 / denom;
  red[t] = m0 * m0 + m1 * m1;
  __syncthreads();
  for (int s = 128; s > 0; s >>= 1) {
    if (t < s) red[t] += red[t + s];
    __syncthreads();
  }
  const float nrm = fmaxf(sqrtf(red[0]), 1e-12f);

  const float* old = prototypes + (size_t)c * EMBED_DIM;
  const float o0 = old[t];
  const float o1 = old[t + 256];
  float u0 = o0, u1 = o1;
  if (cnt > 0) {
    u0 = 0.9f * o0 + 0.1f * (m0 / nrm);
    u1 = 0.9f * o1 + 0.1f * (m1 / nrm);
  }
  outProto[(size_t)c * EMBED_DIM + t]       = u0;
  outProto[(size_t)c * EMBED_DIM + t + 256] = u1;
  if (t == 0) outCounts[c] = class_counts[c] + (float)cnt;
}

extern "C" void kernel_launch(void* const* d_in, const int* in_sizes, int n_in,
                              void* d_out, int out_size, void* d_ws, size_t ws_size,
                              hipStream_t stream) {
  (void)in_sizes; (void)n_in; (void)out_size; (void)ws_size;
  const float* features     = (const float*)d_in[0];
  const float* prototypes   = (const float*)d_in[1];
  const float* class_counts = (const float*)d_in[2];
  const int*   labels       = (const int*)d_in[3];

  float* out      = (float*)d_out;
  float* loss     = out;                                      // [1]
  float* logits   = out + 1;                                  // [BATCH * N_CLASSES]
  float* outProto = logits + (size_t)BATCH * N_CLASSES;       // [N_CLASSES * EMBED_DIM]
  float* outCnt   = outProto + (size_t)N_CLASSES * EMBED_DIM; // [N_CLASSES]

  float* ws      = (float*)d_ws;
  float* featsN  = ws;                                        // BATCH*512
  float* protosT = featsN + (size_t)BATCH * EMBED_DIM;        // 512*1000 (transposed)
  float* rowloss = protosT + (size_t)EMBED_DIM * N_CLASSES;   // BATCH

  rownorm_kernel<<<N_CLASSES, 256, 0, stream>>>(prototypes, protosT, 1, N_CLASSES);
  rownorm_kernel<<<BATCH,     256, 0, stream>>>(features,   featsN,  0, 0);
  gemm_wmma_kernel<<<BATCH / 16, 256, 0, stream>>>(featsN, protosT, logits);
  rowloss_kernel<<<BATCH, 256, 0, stream>>>(logits, labels, rowloss);
  lossreduce_kernel<<<1, 256, 0, stream>>>(rowloss, loss);
  classupdate_kernel<<<N_CLASSES, 256, 0, stream>>>(featsN, labels, prototypes,
                                                    class_counts, outProto, outCnt);
}